// PowerIterationLayer_16913581211844
// MI455X (gfx1250) — compile-verified
//
#include <hip/hip_runtime.h>

typedef float v2f __attribute__((ext_vector_type(2)));
typedef float v8f __attribute__((ext_vector_type(8)));
typedef unsigned int u32x4 __attribute__((ext_vector_type(4)));
typedef int i32x4 __attribute__((ext_vector_type(4)));
typedef int i32x8 __attribute__((ext_vector_type(8)));

#define LDA_S 34                 // padded LDS row stride (32 data + 2 pad dwords)
#define AS_FLOATS (64 * LDA_S)   // A tile: 64 x 32 (stride 34)
#define BT_FLOATS (128 * LDA_S)  // B tile transposed: 128 x 32 (stride 34)
#define AS_OFF_BYTES 0
#define BT_OFF_BYTES (AS_FLOATS * 4)

// ---------------------------------------------------------------------------
// TDM: DMA a 2D tile (rows x 32 floats, row stride = rowStrideElems) from
// global memory into LDS at lds_off, with hardware padding of 2 dwords after
// every 32 dwords -> LDS row stride 34 floats. D# layout per CDNA5 ISA §8.3/8.4.
// Issued by one wave; EXEC ignored by hardware.
// ---------------------------------------------------------------------------
__device__ __forceinline__ void tdm_load_tile_pad34(unsigned lds_off,
                                                    const float* gptr,
                                                    unsigned rows,
                                                    unsigned rowStrideElems)
{
    unsigned long long ga = (unsigned long long)(uintptr_t)gptr;

    // ----- D# group 0: count=1, lds_addr, global_addr, type=2 -----
    u32x4 g0;
    g0.x = 1u;                                        // count=1, user mode
    g0.y = lds_off;                                   // lds_addr (bytes)
    g0.z = (unsigned)(ga & 0xffffffffull);            // global_addr[31:0]
    g0.w = (unsigned)((ga >> 32) & 0x1ffffffull)      // global_addr[56:32]
         | (2u << 30);                                // type = 2 ("image")

    // ----- D# group 1 -----
    const unsigned td0 = 32u, tile0 = 32u;            // 32 floats per row
    const unsigned td1 = rows, tile1 = rows;
    const unsigned long long s0 = (unsigned long long)rowStrideElems;
    // flags: data_size=2 (4B), pad_enable, pad_interval code 4 (=32 dwords),
    //        pad_amount code 1 (=2 dwords)
    const unsigned flags = (2u << 16) | (1u << 20) | (4u << 22) | (1u << 25);
    i32x8 g1;
    g1[0] = (int)flags;
    g1[1] = (int)((td0 & 0xffffu) << 16);                               // tensor_dim0 lo
    g1[2] = (int)(((td0 >> 16) & 0xffffu) | ((td1 & 0xffffu) << 16));   // td0 hi | td1 lo
    g1[3] = (int)(((td1 >> 16) & 0xffffu) | ((tile0 & 0xffffu) << 16)); // td1 hi | tile_dim0
    g1[4] = (int)(tile1 & 0xffffu);                                     // tile_dim1 (tile_dim2=0)
    g1[5] = (int)(unsigned)(s0 & 0xffffffffull);                        // dim0_stride[31:0]
    g1[6] = (int)(unsigned)((s0 >> 32) & 0xffffull);                    // dim0_stride[47:32]
    g1[7] = 0;                                                          // dim1_stride hi = 0

    i32x4 gz4 = (i32x4){0, 0, 0, 0};                  // groups 2/3 unused (2D tensor)
    i32x8 gz8 = (i32x8){0, 0, 0, 0, 0, 0, 0, 0};      // extended group unused
    __builtin_amdgcn_tensor_load_to_lds(g0, g1, gz4, gz4, gz8, 0);
}

// ---------------------------------------------------------------------------
// Generic batched fp32 GEMM: C[b] = op(A[b]) @ op(B[b])  (M x K) x (K x N)
//   TRANSA=1: A stored [K][M]   TRANSB=1: B stored [N][K]
//   MULT_E : multiply result elementwise by E (shape of C) on store
//   BETA   : C += result instead of C = result
// Block = 256 threads (8 wave32), tile 64(M) x 128(N), K staged by 32.
// LDS: As[64][34] (A'[m][k]) and Bt[128][34] (B'[k][n] stored transposed),
// both giving contiguous 8B-aligned b64 fragment loads, bank-conflict-free.
// Layout-preserving tiles are staged by the Tensor Data Mover.
// ---------------------------------------------------------------------------
template<int TRANSA, int TRANSB, int MULT_E, int BETA>
__global__ __launch_bounds__(256)
void gemm_wmma_f32(const float* __restrict__ A, const float* __restrict__ B,
                   const float* __restrict__ E, float* __restrict__ C,
                   int M, int N, int K)
{
    __shared__ float smem[AS_FLOATS + BT_FLOATS];
    float* As = smem;                 // LDS offset 0
    float* Bt = smem + AS_FLOATS;     // LDS offset BT_OFF_BYTES

    const int b      = blockIdx.z;
    const int mBlock = blockIdx.y * 64;
    const int nBlock = blockIdx.x * 128;

    const long aBatch = (long)M * K;
    const long bBatch = (long)K * N;
    const long cBatch = (long)M * N;

    const float* Ab = A + (long)b * aBatch;
    const float* Bb = B + (long)b * bBatch;
    float*       Cb = C + (long)b * cBatch;

    const int tid   = threadIdx.x;
    const int lane  = tid & 31;
    const int wave  = tid >> 5;
    const int waveM = wave >> 2;   // 0..1  -> 2 x 32 = 64 rows
    const int waveN = wave & 3;    // 0..3  -> 4 x 32 = 128 cols
    const int lhi   = lane >> 4;   // 0 or 1 (half-wave)
    const int llo   = lane & 15;

    v8f acc[2][2];
    #pragma unroll
    for (int i = 0; i < 2; ++i)
        #pragma unroll
        for (int j = 0; j < 2; ++j)
            acc[i][j] = (v8f){0.f,0.f,0.f,0.f,0.f,0.f,0.f,0.f};

    for (int k0 = 0; k0 < K; k0 += 32) {
        // ---------------- stage A tile: logical 64 x 32 at (mBlock, k0) ----------------
        if (TRANSA == 0) {
            // layout-preserving copy -> Tensor Data Mover (pad to stride 34)
            if (wave == 0)
                tdm_load_tile_pad34(AS_OFF_BYTES, Ab + (long)mBlock * K + k0, 64u, (unsigned)K);
        } else {
            // A stored [K][M]: load 32 x 64 block, transpose into As[m][k]
            #pragma unroll
            for (int it = 0; it < 2; ++it) {
                int idx  = tid + it * 256;
                int row  = idx >> 4;           // k: 0..31
                int col4 = idx & 15;           // m/4
                float4 vld = *(const float4*)(Ab + (long)(k0 + row) * M + mBlock + col4 * 4);
                As[(col4 * 4 + 0) * LDA_S + row] = vld.x;
                As[(col4 * 4 + 1) * LDA_S + row] = vld.y;
                As[(col4 * 4 + 2) * LDA_S + row] = vld.z;
                As[(col4 * 4 + 3) * LDA_S + row] = vld.w;
            }
        }
        // ---------------- stage B tile: logical 32 x 128 at (k0, nBlock) ---------------
        if (TRANSB == 1) {
            // B stored [N][K]: tile is a direct [n][k] copy -> TDM (pad to stride 34)
            if (wave == 0)
                tdm_load_tile_pad34(BT_OFF_BYTES, Bb + (long)nBlock * K + k0, 128u, (unsigned)K);
        } else {
            // B row-major [K][N]: load 32 x 128, transpose into Bt[n][k]
            #pragma unroll
            for (int it = 0; it < 4; ++it) {
                int idx  = tid + it * 256;
                int row  = idx >> 5;           // k: 0..31
                int col4 = idx & 31;           // n/4
                float4 vld = *(const float4*)(Bb + (long)(k0 + row) * N + nBlock + col4 * 4);
                Bt[(col4 * 4 + 0) * LDA_S + row] = vld.x;
                Bt[(col4 * 4 + 1) * LDA_S + row] = vld.y;
                Bt[(col4 * 4 + 2) * LDA_S + row] = vld.z;
                Bt[(col4 * 4 + 3) * LDA_S + row] = vld.w;
            }
        }
        if (TRANSA == 0 || TRANSB == 1)
            __builtin_amdgcn_s_wait_tensorcnt(0);   // issuing wave drains TDM before barrier
        __syncthreads();

        // ---------------- 8 k-steps of V_WMMA_F32_16X16X4_F32 ----------------
        #pragma unroll
        for (int k4 = 0; k4 < 32; k4 += 4) {
            const int c = k4 + 2 * lhi;            // K pair index for this half-wave
            // A fragment: lane holds A'[m16 + lane%16][c .. c+1]
            v2f afrag[2];
            #pragma unroll
            for (int tm = 0; tm < 2; ++tm) {
                int m = waveM * 32 + tm * 16 + llo;
                afrag[tm] = *(const v2f*)&As[m * LDA_S + c];   // 8B aligned
            }
            // B fragment: lane holds B'[c .. c+1][n16 + lane%16] (Bt contiguous in k)
            v2f bfrag[2];
            #pragma unroll
            for (int tn = 0; tn < 2; ++tn) {
                int n = waveN * 32 + tn * 16 + llo;
                bfrag[tn] = *(const v2f*)&Bt[n * LDA_S + c];   // 8B aligned
            }
            #pragma unroll
            for (int tm = 0; tm < 2; ++tm)
                #pragma unroll
                for (int tn = 0; tn < 2; ++tn)
                    acc[tm][tn] = __builtin_amdgcn_wmma_f32_16x16x4_f32(
                        false, afrag[tm], false, bfrag[tn],
                        (short)0, acc[tm][tn], false, false);
        }
        __syncthreads();
    }

    // ---------------- epilogue: C/D layout = VGPR r -> M = r + 8*lhi, lane%16 -> N ----
    #pragma unroll
    for (int tm = 0; tm < 2; ++tm) {
        #pragma unroll
        for (int tn = 0; tn < 2; ++tn) {
            #pragma unroll
            for (int r = 0; r < 8; ++r) {
                int  gm  = mBlock + waveM * 32 + tm * 16 + r + 8 * lhi;
                int  gn  = nBlock + waveN * 32 + tn * 16 + llo;
                long idx = (long)gm * N + gn;
                float val = acc[tm][tn][r];
                if (BETA)   val += Cb[idx];
                if (MULT_E) val *= E[(long)b * cBatch + idx];
                Cb[idx] = val;
            }
        }
    }
}

// v0 = ones
__global__ void init_ones_kernel(float* __restrict__ v, int n)
{
    int i = blockIdx.x * blockDim.x + threadIdx.x;
    if (i < n) v[i] = 1.0f;
}

// s = 0.5*Zsum + Mp .* v ; per-batch sum(s^2). One block (256 thr) per batch.
__global__ __launch_bounds__(256)
void fuse_reduce_kernel(const float* __restrict__ Zsum, const float* __restrict__ Mp,
                        const float* __restrict__ v, float* __restrict__ S,
                        float* __restrict__ acc)
{
    __shared__ float red[256];
    const int  b    = blockIdx.x;
    const int  n    = 128 * 128;
    const long base = (long)b * n;
    float ss = 0.f;
    for (int i = threadIdx.x; i < n; i += 256) {
        float s = 0.5f * Zsum[base + i] + Mp[base + i] * v[base + i];
        S[base + i] = s;
        ss += s * s;
    }
    red[threadIdx.x] = ss;
    __syncthreads();
    for (int off = 128; off > 0; off >>= 1) {
        if (threadIdx.x < off) red[threadIdx.x] += red[threadIdx.x + off];
        __syncthreads();
    }
    if (threadIdx.x == 0) acc[b] = red[0];
}

// v = S * rsqrt(max(acc[b], 1e-12))
__global__ void scale_kernel(const float* __restrict__ S, const float* __restrict__ acc,
                             float* __restrict__ v)
{
    int i = blockIdx.x * blockDim.x + threadIdx.x;   // 131072 total
    int b = i >> 14;                                  // / (128*128)
    float scale = rsqrtf(fmaxf(acc[b], 1e-12f));
    v[i] = S[i] * scale;
}

extern "C" void kernel_launch(void* const* d_in, const int* in_sizes, int n_in,
                              void* d_out, int out_size, void* d_ws, size_t ws_size,
                              hipStream_t stream)
{
    const float* Mp = (const float*)d_in[0];   // [8,128,128]
    const float* Mq = (const float*)d_in[1];   // [8,512,512]
    const float* G1 = (const float*)d_in[2];   // [8,128,512]
    const float* G2 = (const float*)d_in[3];   // [8,128,512]
    const float* H1 = (const float*)d_in[4];   // [8,128,512]
    const float* H2 = (const float*)d_in[5];   // [8,128,512]
    float* v = (float*)d_out;                  // [8,128,128] — live iterate

    const int Bn = 8, n1 = 128, n2 = 128, e1 = 512, e2 = 512;
    const int NV = Bn * n1 * n2;               // 131072

    float* ws = (float*)d_ws;
    size_t off = 0;
    float* T1   = ws + off; off += (size_t)Bn * e1 * n2;   // H1^T v
    float* T2   = ws + off; off += (size_t)Bn * e1 * n2;   // G1^T v
    float* Y1   = ws + off; off += (size_t)Bn * e1 * e2;   // Mq .* (T1 H2)
    float* Y2   = ws + off; off += (size_t)Bn * e1 * e2;   // Mq .* (T2 G2)
    float* Z1a  = ws + off; off += (size_t)Bn * n1 * e2;   // G1 Y1
    float* Z1b  = ws + off; off += (size_t)Bn * n1 * e2;   // H1 Y2
    float* Zs   = ws + off; off += (size_t)Bn * n1 * n2;   // Z1a G2^T + Z1b H2^T
    float* S    = ws + off; off += (size_t)Bn * n1 * n2;
    float* accv = ws + off; off += Bn;

    dim3 blk(256);

    init_ones_kernel<<<dim3((NV + 255) / 256), blk, 0, stream>>>(v, NV);

    for (int it = 0; it < 30; ++it) {
        // T = Q1^T @ v : M=e1, N=n2, K=n1  (call1 Q1=H1, call2 Q1=G1)
        gemm_wmma_f32<1,0,0,0><<<dim3(n2/128, e1/64, Bn), blk, 0, stream>>>(H1, v, nullptr, T1, e1, n2, n1);
        gemm_wmma_f32<1,0,0,0><<<dim3(n2/128, e1/64, Bn), blk, 0, stream>>>(G1, v, nullptr, T2, e1, n2, n1);
        // Y = Mq .* (T @ Q2) : M=e1, N=e2, K=n2
        gemm_wmma_f32<0,0,1,0><<<dim3(e2/128, e1/64, Bn), blk, 0, stream>>>(T1, H2, Mq, Y1, e1, e2, n2);
        gemm_wmma_f32<0,0,1,0><<<dim3(e2/128, e1/64, Bn), blk, 0, stream>>>(T2, G2, Mq, Y2, e1, e2, n2);
        // Z1 = P1 @ Y : M=n1, N=e2, K=e1
        gemm_wmma_f32<0,0,0,0><<<dim3(e2/128, n1/64, Bn), blk, 0, stream>>>(G1, Y1, nullptr, Z1a, n1, e2, e1);
        gemm_wmma_f32<0,0,0,0><<<dim3(e2/128, n1/64, Bn), blk, 0, stream>>>(H1, Y2, nullptr, Z1b, n1, e2, e1);
        // Zs = Z1a @ G2^T + Z1b @ H2^T : M=n1, N=n2, K=e2 (B stored [N][K] -> transB, TDM path)
        gemm_wmma_f32<0,1,0,0><<<dim3(n2/128, n1/64, Bn), blk, 0, stream>>>(Z1a, G2, nullptr, Zs, n1, n2, e2);
        gemm_wmma_f32<0,1,0,1><<<dim3(n2/128, n1/64, Bn), blk, 0, stream>>>(Z1b, H2, nullptr, Zs, n1, n2, e2);
        // u = l2norm(Zs/2 + Mp .* v)  (reduction then scale; scale writes v)
        fuse_reduce_kernel<<<dim3(Bn), blk, 0, stream>>>(Zs, Mp, v, S, accv);
        scale_kernel<<<dim3(NV / 256), blk, 0, stream>>>(S, accv, v);
    }
}